// Model_31001073942588
// MI455X (gfx1250) — compile-verified
//
#include <hip/hip_runtime.h>
#include <math.h>

typedef __attribute__((ext_vector_type(2))) float v2f;
typedef __attribute__((ext_vector_type(8))) float v8f;

static constexpr int Bsz  = 1024;   // batch
static constexpr int NIFc = 2048;   // image features
static constexpr int Vc   = 10000;  // vocab
static constexpr int Ec   = 512;    // embedding
static constexpr int Hc   = 1024;   // hidden
static constexpr int Lc   = 32;     // sentence length

// ---------------- WMMA helper (fp32, 16x16x4) ----------------
__device__ __forceinline__ v8f wmma4(v2f a, v2f b, v8f c) {
  // D = A(16x4 f32) * B(4x16 f32) + C(16x16 f32)
  return __builtin_amdgcn_wmma_f32_16x16x4_f32(false, a, false, b, (short)0, c,
                                               false, false);
}

// 2x2 register-blocked K-loop: 4 loads -> 4 WMMAs per K-step (1 load/WMMA).
// Row pointers are this lane's rows, pre-offset by 2*khalf; all K-contiguous.
struct Acc4 { v8f c00, c01, c10, c11; };

__device__ __forceinline__ void mm_nt_2x2(const float* __restrict__ A0,
                                          const float* __restrict__ A1,
                                          const float* __restrict__ B0,
                                          const float* __restrict__ B1,
                                          int K, Acc4& acc) {
#pragma unroll 2
  for (int k = 0; k < K; k += 4) {
    v2f a0 = *(const v2f*)(A0 + k);
    v2f a1 = *(const v2f*)(A1 + k);
    v2f b0 = *(const v2f*)(B0 + k);
    v2f b1 = *(const v2f*)(B1 + k);
    acc.c00 = wmma4(a0, b0, acc.c00);
    acc.c01 = wmma4(a0, b1, acc.c01);
    acc.c10 = wmma4(a1, b0, acc.c10);
    acc.c11 = wmma4(a1, b1, acc.c11);
  }
}

__device__ __forceinline__ void store_tile(float* __restrict__ C, int N, int m0,
                                           int n, v8f acc, float bn, int half) {
  float* Cb = C + (size_t)(m0 + 8 * half) * N + n;
#pragma unroll
  for (int r = 0; r < 8; ++r) Cb[(size_t)r * N] = acc[r] + bn;
}

// ---------------- GEMM: C[M,N] = A[M,K] * Bt[N,K]^T + bias[N] ----------------
// One wave computes a 32x32 output block (2x2 WMMA tiles).
__global__ void gemm_nt_bias(const float* __restrict__ A, const float* __restrict__ Bt,
                             const float* __restrict__ bias, float* __restrict__ C,
                             int M, int N, int K) {
  int wave    = blockIdx.x * (blockDim.x >> 5) + (threadIdx.x >> 5);
  int tilesN  = N >> 4;
  int tilesN2 = (tilesN + 1) >> 1;
  int tm2 = wave / tilesN2, tn2 = wave % tilesN2;
  if (tm2 * 32 >= M) return;                 // wave-uniform: EXEC stays full
  int lane = threadIdx.x & 31;
  int half = lane >> 4;
  int mn   = lane & 15;
  int ntile0 = tn2 * 2;
  int ntile1 = (ntile0 + 1 < tilesN) ? (ntile0 + 1) : (tilesN - 1); // clamp tail
  const float* A0 = A  + (size_t)(tm2 * 32 + mn) * K + 2 * half;
  const float* A1 = A0 + (size_t)16 * K;
  const float* B0 = Bt + (size_t)(ntile0 * 16 + mn) * K + 2 * half;
  const float* B1 = Bt + (size_t)(ntile1 * 16 + mn) * K + 2 * half;
  Acc4 acc = {};
  mm_nt_2x2(A0, A1, B0, B1, K, acc);
  float bn0 = bias ? bias[ntile0 * 16 + mn] : 0.f;
  float bn1 = bias ? bias[ntile1 * 16 + mn] : 0.f;
  int n0 = ntile0 * 16 + mn, n1 = ntile1 * 16 + mn;
  store_tile(C, N, tm2 * 32,      n0, acc.c00, bn0, half);
  store_tile(C, N, tm2 * 32 + 16, n0, acc.c10, bn0, half);
  if (ntile0 + 1 < tilesN) {                 // wave-uniform store guard
    store_tile(C, N, tm2 * 32,      n1, acc.c01, bn1, half);
    store_tile(C, N, tm2 * 32 + 16, n1, acc.c11, bn1, half);
  }
}

// ---- fused LSTM gates: G[B,4H] = X*Wih^T + Hs*Whh^T + bih + bhh ----
__global__ void lstm_gates(const float* __restrict__ X,  const float* __restrict__ Wih,
                           const float* __restrict__ Hs, const float* __restrict__ Whh,
                           const float* __restrict__ bih, const float* __restrict__ bhh,
                           float* __restrict__ G, int Kx, int Kh) {
  const int N = 4 * Hc;                       // 4096 -> 256 tiles (even)
  int wave    = blockIdx.x * (blockDim.x >> 5) + (threadIdx.x >> 5);
  int tilesN2 = (N >> 4) >> 1;
  int tm2 = wave / tilesN2, tn2 = wave % tilesN2;
  if (tm2 * 32 >= Bsz) return;
  int lane = threadIdx.x & 31;
  int half = lane >> 4;
  int mn   = lane & 15;
  int n0 = tn2 * 32 + mn, n1 = n0 + 16;
  Acc4 acc = {};
  {
    const float* A0 = X + (size_t)(tm2 * 32 + mn) * Kx + 2 * half;
    const float* B0 = Wih + (size_t)(tn2 * 32 + mn) * Kx + 2 * half;
    mm_nt_2x2(A0, A0 + (size_t)16 * Kx, B0, B0 + (size_t)16 * Kx, Kx, acc);
  }
  {
    const float* A0 = Hs + (size_t)(tm2 * 32 + mn) * Kh + 2 * half;
    const float* B0 = Whh + (size_t)(tn2 * 32 + mn) * Kh + 2 * half;
    mm_nt_2x2(A0, A0 + (size_t)16 * Kh, B0, B0 + (size_t)16 * Kh, Kh, acc);
  }
  float bn0 = bih[n0] + bhh[n0];
  float bn1 = bih[n1] + bhh[n1];
  store_tile(G, N, tm2 * 32,      n0, acc.c00, bn0, half);
  store_tile(G, N, tm2 * 32 + 16, n0, acc.c10, bn0, half);
  store_tile(G, N, tm2 * 32,      n1, acc.c01, bn1, half);
  store_tile(G, N, tm2 * 32 + 16, n1, acc.c11, bn1, half);
}

// ---- LSTM cell elementwise: gates ordered i, f, g, o ----
__global__ void lstm_update(const float* __restrict__ G, float* __restrict__ h,
                            float* __restrict__ c) {
  int i = blockIdx.x * blockDim.x + threadIdx.x;
  if (i >= Bsz * Hc) return;
  int b = i / Hc, j = i % Hc;
  const float* g = G + (size_t)b * 4 * Hc;
  float ig = g[j], fg = g[Hc + j], gg = g[2 * Hc + j], og = g[3 * Hc + j];
  float si = 1.f / (1.f + expf(-ig));
  float sf = 1.f / (1.f + expf(-fg));
  float so = 1.f / (1.f + expf(-og));
  float cn = sf * c[i] + si * tanhf(gg);
  c[i] = cn;
  h[i] = so * tanhf(cn);
}

// ---- argmax + log_softmax(logp of argmax) + embedding gather, one block/row ----
__global__ void argmax_logp_gather(const float* __restrict__ logits,
                                   const float* __restrict__ emb,
                                   int* __restrict__ msg_t, float* __restrict__ logp,
                                   float* __restrict__ Xnext) {
  int b = blockIdx.x;
  const float* row = logits + (size_t)b * Vc;
  __shared__ float smx[256];
  __shared__ int   sid[256];
  __shared__ float ssm[256];
  int t = threadIdx.x;
  float mx = -__builtin_inff();
  int   mi = 0x7fffffff;
  for (int v = t; v < Vc; v += 256) {
    float x = row[v];
    if (x > mx || (x == mx && v < mi)) { mx = x; mi = v; }
  }
  smx[t] = mx; sid[t] = mi;
  __syncthreads();
  for (int s = 128; s > 0; s >>= 1) {
    if (t < s) {
      float ox = smx[t + s]; int oi = sid[t + s];
      if (ox > smx[t] || (ox == smx[t] && oi < sid[t])) { smx[t] = ox; sid[t] = oi; }
    }
    __syncthreads();
  }
  float gmax = smx[0];
  int   gidx = sid[0];
  float acc = 0.f;
  for (int v = t; v < Vc; v += 256) acc += expf(row[v] - gmax);
  ssm[t] = acc;
  __syncthreads();
  for (int s = 128; s > 0; s >>= 1) {
    if (t < s) ssm[t] += ssm[t + s];
    __syncthreads();
  }
  if (t == 0) {
    msg_t[b] = gidx;
    logp[b]  = row[gidx] - gmax - logf(ssm[0]);
  }
  const float* e = emb + (size_t)gidx * Ec;
  for (int j = t; j < Ec; j += 256) Xnext[(size_t)b * Ec + j] = e[j];
}

// ---- embedding gather for receiver ----
__global__ void gather_rows(const int* __restrict__ idx, const float* __restrict__ emb,
                            float* __restrict__ X) {
  int b = blockIdx.x;
  const float* e = emb + (size_t)idx[b] * Ec;
  for (int j = threadIdx.x; j < Ec; j += blockDim.x) X[(size_t)b * Ec + j] = e[j];
}

// ---- broadcast start-token embedding ----
__global__ void set_start_x(const int* __restrict__ sidx, const float* __restrict__ emb,
                            float* __restrict__ X) {
  int b = blockIdx.x;
  const float* e = emb + (size_t)(*sidx) * Ec;
  for (int j = threadIdx.x; j < Ec; j += blockDim.x) X[(size_t)b * Ec + j] = e[j];
}

__global__ void zero_f32(float* __restrict__ p, int n) {
  int i = blockIdx.x * blockDim.x + threadIdx.x;
  if (i < n) p[i] = 0.f;
}

// ---- fused hinge loss: wave computes a 16x16 (i,j) tile via 4 WMMA chains ----
__global__ void hinge_loss(const float* __restrict__ target, const float* __restrict__ dist,
                           const float* __restrict__ R, const float* __restrict__ logp,
                           float* __restrict__ out) {
  int wave   = blockIdx.x * (blockDim.x >> 5) + (threadIdx.x >> 5);
  const int tilesJ = Bsz >> 4;
  int ti = wave / tilesJ, tj = wave % tilesJ;
  if (ti * 16 >= Bsz) return;
  int lane = threadIdx.x & 31;
  int half = lane >> 4;
  int mn   = lane & 15;
  const float* Brow = R + (size_t)(tj * 16 + mn) * NIFc + 2 * half;   // r[j,:]
  const float* At  = target + (size_t)(ti * 16 + mn) * NIFc + 2 * half;
  const float* Ad0 = dist + ((size_t)0 * Bsz + ti * 16 + mn) * NIFc + 2 * half;
  const float* Ad1 = dist + ((size_t)1 * Bsz + ti * 16 + mn) * NIFc + 2 * half;
  const float* Ad2 = dist + ((size_t)2 * Bsz + ti * 16 + mn) * NIFc + 2 * half;
  v8f at = {}, a0 = {}, a1 = {}, a2 = {};
  for (int k = 0; k < NIFc; k += 4) {
    v2f bv = *(const v2f*)(Brow + k);
    at = wmma4(*(const v2f*)(At + k),  bv, at);
    a0 = wmma4(*(const v2f*)(Ad0 + k), bv, a0);
    a1 = wmma4(*(const v2f*)(Ad1 + k), bv, a1);
    a2 = wmma4(*(const v2f*)(Ad2 + k), bv, a2);
  }
  float lp   = logp[tj * 16 + mn];     // log_prob broadcasts over the LAST dim (j)
  float part = 0.f;
#pragma unroll
  for (int r = 0; r < 8; ++r) {
    float tv = at[r];
    float l  = fmaxf(0.f, 1.f - tv + a0[r]) +
               fmaxf(0.f, 1.f - tv + a1[r]) +
               fmaxf(0.f, 1.f - tv + a2[r]);
    part += -l * lp;
  }
  for (int off = 16; off > 0; off >>= 1) part += __shfl_xor(part, off, 32);
  if (lane == 0) atomicAdd(out, part * (1.f / ((float)Bsz * (float)Bsz)));
}

// ---------------- host side ----------------
static inline int gemm_grid_2x2(int M, int N) {
  int tilesN2 = ((N / 16) + 1) / 2;
  int tiles   = (M / 32) * tilesN2;   // 8 waves per 256-thread block
  return (tiles + 7) / 8;
}

extern "C" void kernel_launch(void* const* d_in, const int* in_sizes, int n_in,
                              void* d_out, int out_size, void* d_ws, size_t ws_size,
                              hipStream_t stream) {
  (void)in_sizes; (void)n_in; (void)out_size; (void)ws_size;
  const float* target = (const float*)d_in[0];
  const float* dist   = (const float*)d_in[1];
  const int*   sidx   = (const int*)d_in[2];
  // d_in[3] = max_sentence_length (compile-time Lc)
  const float* W_aff  = (const float*)d_in[4];
  const float* b_aff  = (const float*)d_in[5];
  const float* emb_s  = (const float*)d_in[6];
  const float* Wih_s  = (const float*)d_in[7];
  const float* Whh_s  = (const float*)d_in[8];
  const float* bih_s  = (const float*)d_in[9];
  const float* bhh_s  = (const float*)d_in[10];
  const float* Wp     = (const float*)d_in[11];
  const float* bp     = (const float*)d_in[12];
  const float* emb_r  = (const float*)d_in[13];
  const float* Wih_r  = (const float*)d_in[14];
  const float* Whh_r  = (const float*)d_in[15];
  const float* bih_r  = (const float*)d_in[16];
  const float* bhh_r  = (const float*)d_in[17];
  const float* W_out  = (const float*)d_in[18];
  const float* b_out  = (const float*)d_in[19];
  float* out = (float*)d_out;

  // workspace arena (floats), 64-element aligned slots
  float* W = (float*)d_ws;
  size_t off = 0;
  auto take = [&](size_t n) { float* p = W + off; off += (n + 63) & ~(size_t)63; return p; };
  float* h_s    = take((size_t)Bsz * Hc);
  float* c_s    = take((size_t)Bsz * Hc);
  float* x_s    = take((size_t)Bsz * Ec);
  float* gates  = take((size_t)Bsz * 4 * Hc);
  float* logits = take((size_t)Bsz * Vc);
  float* logp   = take((size_t)Bsz);
  float* h_r    = take((size_t)Bsz * Hc);
  float* c_r    = take((size_t)Bsz * Hc);
  float* x_r    = take((size_t)Bsz * Ec);
  float* rbuf   = take((size_t)Bsz * NIFc);
  int*   msg    = (int*)take((size_t)Lc * Bsz);

  const dim3 blk(256);

  // init: c0=0, receiver h/c=0, x0 = emb_s[start_token]
  zero_f32<<<(Bsz * Hc + 255) / 256, blk, 0, stream>>>(c_s, Bsz * Hc);
  zero_f32<<<(Bsz * Hc + 255) / 256, blk, 0, stream>>>(h_r, Bsz * Hc);
  zero_f32<<<(Bsz * Hc + 255) / 256, blk, 0, stream>>>(c_r, Bsz * Hc);
  set_start_x<<<Bsz, blk, 0, stream>>>(sidx, emb_s, x_s);

  // h0 = target @ W_aff^T + b_aff
  gemm_nt_bias<<<gemm_grid_2x2(Bsz, Hc), blk, 0, stream>>>(target, W_aff, b_aff, h_s,
                                                           Bsz, Hc, NIFc);

  // ---- Sender: greedy decode ----
  for (int t = 0; t < Lc; ++t) {
    lstm_gates<<<gemm_grid_2x2(Bsz, 4 * Hc), blk, 0, stream>>>(x_s, Wih_s, h_s, Whh_s,
                                                               bih_s, bhh_s, gates,
                                                               Ec, Hc);
    lstm_update<<<(Bsz * Hc + 255) / 256, blk, 0, stream>>>(gates, h_s, c_s);
    gemm_nt_bias<<<gemm_grid_2x2(Bsz, Vc), blk, 0, stream>>>(h_s, Wp, bp, logits,
                                                             Bsz, Vc, Hc);
    argmax_logp_gather<<<Bsz, blk, 0, stream>>>(logits, emb_s, msg + (size_t)t * Bsz,
                                                logp, x_s);
  }

  // ---- Receiver: encode message ----
  for (int t = 0; t < Lc; ++t) {
    gather_rows<<<Bsz, blk, 0, stream>>>(msg + (size_t)t * Bsz, emb_r, x_r);
    lstm_gates<<<gemm_grid_2x2(Bsz, 4 * Hc), blk, 0, stream>>>(x_r, Wih_r, h_r, Whh_r,
                                                               bih_r, bhh_r, gates,
                                                               Ec, Hc);
    lstm_update<<<(Bsz * Hc + 255) / 256, blk, 0, stream>>>(gates, h_r, c_r);
  }

  // r = h_r @ W_out^T + b_out
  gemm_nt_bias<<<gemm_grid_2x2(Bsz, NIFc), blk, 0, stream>>>(h_r, W_out, b_out, rbuf,
                                                             Bsz, NIFc, Hc);

  // loss
  zero_f32<<<1, 32, 0, stream>>>(out, 1);
  hinge_loss<<<(((Bsz / 16) * (Bsz / 16)) + 7) / 8, blk, 0, stream>>>(target, dist,
                                                                      rbuf, logp, out);
}